// SAGE_64141041599038
// MI455X (gfx1250) — compile-verified
//
#include <hip/hip_runtime.h>
#include <hip/hip_bf16.h>

// ---------------------------------------------------------------------------
// GraphSAGE 2-layer forward for MI455X (gfx1250, wave32).
//   L1: h = relu(((scatter_sum(x,src->dst) + x) / (deg+1)) @ W1n + b1)
//   L2: out = h @ W2s + b2 + (scatter_sum(h,src->dst)/max(deg,1)) @ W2n
// Aggregation via fp32 global atomics (x/h/accum all L2-resident: ~155 MB
// total vs 192 MB global L2, so RMW traffic never leaves L2).
// GEMMs via V_WMMA_F32_16X16X4_F32 (full fp32 precision, CDNA5 matrix pipe).
// ---------------------------------------------------------------------------

typedef float v2f __attribute__((ext_vector_type(2)));
typedef float v8f __attribute__((ext_vector_type(8)));

#define IN_F   128
#define HID_F  128
#define OUT_F  64

// ---------------- degree: deg[dst] += 1.0 ----------------------------------
__global__ void sage_deg_kernel(const int* __restrict__ ei, float* __restrict__ deg, int E) {
    int e = blockIdx.x * blockDim.x + threadIdx.x;
    if (e < E) atomicAdd(&deg[ei[E + e]], 1.0f);
}

// ---------------- scatter-add: accum[dst,:] += feat[src,:] -----------------
// one thread per (edge, 4-float chunk); 32 chunks per 128-wide row.
__global__ void sage_scatter_kernel(const int* __restrict__ ei,
                                    const float* __restrict__ feat,
                                    float* __restrict__ accum, int E) {
    long long idx   = (long long)blockIdx.x * blockDim.x + threadIdx.x;
    long long total = (long long)E * 32;
    if (idx >= total) return;
    int e  = (int)(idx >> 5);
    int c4 = (int)(idx & 31) * 4;
    int s  = ei[e];
    int d  = ei[E + e];
    const float4 v = *(const float4*)&feat[(size_t)s * 128 + c4];
    float* ap = &accum[(size_t)d * 128 + c4];
    atomicAdd(ap + 0, v.x);
    atomicAdd(ap + 1, v.y);
    atomicAdd(ap + 2, v.z);
    atomicAdd(ap + 3, v.w);
}

// ---------------- layer 1 GEMM: relu(A @ W1n + b1) -------------------------
// A = (accum + x) / (deg + 1), staged per 16-row tile in LDS.
// 128 threads = 4 waves; each wave computes two 16x16 output tiles (128 cols).
__global__ __launch_bounds__(128) void sage_gemm1_kernel(
        const float* __restrict__ x, const float* __restrict__ W1,
        const float* __restrict__ b1, const float* __restrict__ deg,
        const float* __restrict__ accum, float* __restrict__ h, int N) {
    __shared__ float As[16 * IN_F];

    const int row0 = blockIdx.x * 16;
    const int tid  = threadIdx.x;

    // Stage preprocessed A tile: 16 rows x 128 cols (uniform loop, no exec churn)
    for (int i = tid; i < 16 * IN_F; i += 128) {
        int r = i >> 7, c = i & 127;
        int g = row0 + r;
        if (g >= N) g = N - 1;               // clamp (N is a multiple of 16 anyway)
        float inv = 1.0f / (deg[g] + 1.0f);
        As[i] = (accum[(size_t)g * 128 + c] + x[(size_t)g * 128 + c]) * inv;
    }
    __syncthreads();

    const int  wave  = tid >> 5;
    const int  lane  = tid & 31;
    const int  lhalf = lane & 15;
    const int  khi   = (lane >= 16) ? 2 : 0;   // K offset for upper lane half
    const int  mhi   = (lane >= 16) ? 8 : 0;   // M offset for C/D rows
    const bool full  = (row0 + 16) <= N;       // uniform per block

    for (int t = 0; t < 2; ++t) {
        const int cn = wave * 32 + t * 16;     // column tile base
        // one unconditional prefetch of this wave's weight panel
        // (global_prefetch_b8; W is L2-hot, no branch overhead)
        __builtin_prefetch(&W1[(size_t)64 * HID_F + cn + lhalf], 0, 3);
        v8f c = {};
        #pragma unroll 4
        for (int kk = 0; kk < IN_F / 4; ++kk) {
            const int k0 = kk * 4 + khi;
            v2f a, b;
            a.x = As[lhalf * IN_F + k0];
            a.y = As[lhalf * IN_F + k0 + 1];
            b.x = W1[(size_t)(k0)     * HID_F + cn + lhalf];
            b.y = W1[(size_t)(k0 + 1) * HID_F + cn + lhalf];
            c = __builtin_amdgcn_wmma_f32_16x16x4_f32(
                    false, a, false, b, (short)0, c, false, false);
        }
        const float bias = b1[cn + lhalf];
        float* hp = &h[(size_t)(row0 + mhi) * HID_F + cn + lhalf];
        if (full) {
            // hot path: unguarded store clause with constant byte offsets
            #pragma unroll
            for (int r = 0; r < 8; ++r) {
                float v = c[r] + bias;
                hp[(size_t)r * HID_F] = v > 0.0f ? v : 0.0f;
            }
        } else {
            #pragma unroll
            for (int r = 0; r < 8; ++r) {
                if (row0 + mhi + r < N) {
                    float v = c[r] + bias;
                    hp[(size_t)r * HID_F] = v > 0.0f ? v : 0.0f;
                }
            }
        }
    }
}

// ---------------- layer 2 GEMM: h @ W2s + b2 + mean @ W2n ------------------
// Stages both A matrices (h rows and accum/max(deg,1) rows) in LDS.
// 4 waves x one 16-col tile each = 64 output cols.
__global__ __launch_bounds__(128) void sage_gemm2_kernel(
        const float* __restrict__ h, const float* __restrict__ W2s,
        const float* __restrict__ W2n, const float* __restrict__ b2,
        const float* __restrict__ deg, const float* __restrict__ accum,
        float* __restrict__ out, int N) {
    __shared__ float Hs[16 * HID_F];
    __shared__ float Ms[16 * HID_F];

    const int row0 = blockIdx.x * 16;
    const int tid  = threadIdx.x;

    for (int i = tid; i < 16 * HID_F; i += 128) {
        int r = i >> 7, c = i & 127;
        int g = row0 + r;
        if (g >= N) g = N - 1;
        float d = deg[g];
        d = d > 1.0f ? d : 1.0f;
        Hs[i] = h[(size_t)g * 128 + c];
        Ms[i] = accum[(size_t)g * 128 + c] / d;
    }
    __syncthreads();

    const int  wave  = tid >> 5;
    const int  lane  = tid & 31;
    const int  lhalf = lane & 15;
    const int  khi   = (lane >= 16) ? 2 : 0;
    const int  mhi   = (lane >= 16) ? 8 : 0;
    const int  cn    = wave * 16;              // 4 waves x 16 cols = 64
    const bool full  = (row0 + 16) <= N;

    __builtin_prefetch(&W2n[(size_t)cn + lhalf], 0, 3);

    v8f c = {};
    // fc_self: h @ W2s
    #pragma unroll 4
    for (int kk = 0; kk < HID_F / 4; ++kk) {
        const int k0 = kk * 4 + khi;
        v2f a, b;
        a.x = Hs[lhalf * HID_F + k0];
        a.y = Hs[lhalf * HID_F + k0 + 1];
        b.x = W2s[(size_t)(k0)     * OUT_F + cn + lhalf];
        b.y = W2s[(size_t)(k0 + 1) * OUT_F + cn + lhalf];
        c = __builtin_amdgcn_wmma_f32_16x16x4_f32(
                false, a, false, b, (short)0, c, false, false);
    }
    // fc_neigh: mean @ W2n, accumulated into the same C
    #pragma unroll 4
    for (int kk = 0; kk < HID_F / 4; ++kk) {
        const int k0 = kk * 4 + khi;
        v2f a, b;
        a.x = Ms[lhalf * HID_F + k0];
        a.y = Ms[lhalf * HID_F + k0 + 1];
        b.x = W2n[(size_t)(k0)     * OUT_F + cn + lhalf];
        b.y = W2n[(size_t)(k0 + 1) * OUT_F + cn + lhalf];
        c = __builtin_amdgcn_wmma_f32_16x16x4_f32(
                false, a, false, b, (short)0, c, false, false);
    }

    const float bias = b2[cn + lhalf];
    float* op = &out[(size_t)(row0 + mhi) * OUT_F + cn + lhalf];
    if (full) {
        #pragma unroll
        for (int r = 0; r < 8; ++r)
            op[(size_t)r * OUT_F] = c[r] + bias;
    } else {
        #pragma unroll
        for (int r = 0; r < 8; ++r)
            if (row0 + mhi + r < N)
                op[(size_t)r * OUT_F] = c[r] + bias;
    }
}

// ---------------------------------------------------------------------------
static inline size_t align256(size_t v) { return (v + 255) & ~(size_t)255; }

extern "C" void kernel_launch(void* const* d_in, const int* in_sizes, int n_in,
                              void* d_out, int out_size, void* d_ws, size_t ws_size,
                              hipStream_t stream) {
    const float* x   = (const float*)d_in[0];
    const float* W1n = (const float*)d_in[1];
    const float* b1  = (const float*)d_in[2];
    const float* W2n = (const float*)d_in[3];
    const float* W2s = (const float*)d_in[4];
    const float* b2  = (const float*)d_in[5];
    const int*   ei  = (const int*)d_in[6];

    const int N = in_sizes[0] / IN_F;   // 100000
    const int E = in_sizes[6] / 2;      // 1.6M

    // workspace layout: deg[N] | accum[N*128] | h[N*128]   (~103 MB)
    char*  ws    = (char*)d_ws;
    float* deg   = (float*)ws;
    float* accum = (float*)(ws + align256((size_t)N * sizeof(float)));
    float* h     = (float*)(ws + align256((size_t)N * sizeof(float))
                               + (size_t)N * HID_F * sizeof(float));

    hipMemsetAsync(deg,   0, (size_t)N * sizeof(float), stream);
    hipMemsetAsync(accum, 0, (size_t)N * IN_F * sizeof(float), stream);

    // degrees
    sage_deg_kernel<<<(E + 255) / 256, 256, 0, stream>>>(ei, deg, E);

    // layer-1 scatter: accum += x[src]
    const long long chunks = (long long)E * 32;
    const int sgrid = (int)((chunks + 255) / 256);
    sage_scatter_kernel<<<sgrid, 256, 0, stream>>>(ei, x, accum, E);

    // layer-1 GEMM + relu -> h
    const int tiles = (N + 15) / 16;
    sage_gemm1_kernel<<<tiles, 128, 0, stream>>>(x, W1n, b1, deg, accum, h, N);

    // layer-2 scatter: accum = 0; accum += h[src]
    hipMemsetAsync(accum, 0, (size_t)N * HID_F * sizeof(float), stream);
    sage_scatter_kernel<<<sgrid, 256, 0, stream>>>(ei, h, accum, E);

    // layer-2 dual GEMM -> out
    sage_gemm2_kernel<<<tiles, 128, 0, stream>>>(h, W2s, W2n, b2, deg, accum,
                                                 (float*)d_out, N);
}